// FixedKAN_70738111365140
// MI455X (gfx1250) — compile-verified
//
#include <hip/hip_runtime.h>

typedef __attribute__((ext_vector_type(2))) float v2f;
typedef __attribute__((ext_vector_type(8))) float v8f;

// ---------------------------------------------------------------------------
// Kernel 1: column-sum reduction of the three coefficient matrices via WMMA.
// colsum[d] = sum_w C[w][d]  computed as  ones(16x4) x C_slice(4x16), chained
// accumulation over row-blocks of 4. One wave (32 lanes), EXEC all ones.
//
// B-matrix (4x16 f32, 2 VGPRs) layout per CDNA5 ISA 7.12.2: row striped
// across lanes within a VGPR; column index = lane % 16; the 4 K-rows occupy
// the (VGPR, lane-half) slots. Since A is all-ones, D[m,n] = sum_k B[k,n],
// which is invariant to which K maps to which slot — we only need each slot
// of column d to carry a distinct row w of C[:,d].
//
// All masking is branch-free (bitwise bools -> v_cndmask), so EXEC stays
// all-ones through the whole WMMA loop and both loads issue unconditionally
// at clamped in-bounds addresses.
// ---------------------------------------------------------------------------
__global__ __launch_bounds__(32) void kan_colsum_wmma(
    const float* __restrict__ c0, const float* __restrict__ c1,
    const float* __restrict__ c2, int r0, int r1, int r2,
    float* __restrict__ sums /* 24 floats: s0[8], s1[8], s2[8] */)
{
    const int lane = threadIdx.x;          // 0..31
    const int d    = lane & 15;            // column (degree); cols 8..15 unused
    const int half = lane >> 4;            // lane half selects 2 of the 4 K-rows
    const int dc   = d & 7;                // clamped degree for in-bounds loads
    const int dvalid = (int)(d < 8);       // non-short-circuit mask

    v2f ones;
    ones.x = 1.0f;
    ones.y = 1.0f;

    const float* mats[3] = {c0, c1, c2};
    const int    rows[3] = {r0, r1, r2};
    v8f acc[3];

    #pragma unroll
    for (int l = 0; l < 3; ++l) {
        const float* C = mats[l];
        const int R = rows[l];
        const int iters = (R + 3) >> 2;    // row-blocks of 4 (K=4 per WMMA)
        v8f a = {};                        // zero accumulator
        for (int it = 0; it < iters; ++it) {
            const int w0 = it * 4 + half * 2;
            const int w1 = w0 + 1;
            const int m0 = dvalid & (int)(w0 < R);   // bitwise: no branches
            const int m1 = dvalid & (int)(w1 < R);
            // Unconditional loads at clamped addresses; selects zero out
            // invalid slots. EXEC untouched.
            const float b0 = C[(m0 ? w0 : 0) * 8 + dc];
            const float b1 = C[(m1 ? w1 : 0) * 8 + dc];
            v2f b;
            b.x = m0 ? b0 : 0.0f;
            b.y = m1 ? b1 : 0.0f;
            a = __builtin_amdgcn_wmma_f32_16x16x4_f32(
                    /*neg_a=*/false, ones, /*neg_b=*/false, b,
                    /*c_mod=*/(short)0, a, /*reuse_a=*/false, /*reuse_b=*/false);
        }
        acc[l] = a;
    }

    // C/D layout: lane j (j<16), VGPR0 holds D[M=0, N=j]; every M row holds
    // the same colsum, so lane d's acc[l][0] == colsum_l[d].
    if (lane < 8) {
        sums[lane]      = acc[0][0];
        sums[8 + lane]  = acc[1][0];
        sums[16 + lane] = acc[2][0];
    }
}

// ---------------------------------------------------------------------------
// Clenshaw evaluation of sum_{d=0..7} s_d * T_d(x): 7 fused ops, minimal VALU
// so the streaming kernel stays HBM-bound (32 MB total -> ~1.4 us floor at
// 23.3 TB/s).
// ---------------------------------------------------------------------------
__device__ __forceinline__ float cheb_eval8(float x, const float s[8]) {
    const float t  = x + x;
    const float b7 = s[7];
    const float b6 = fmaf(t, b7, s[6]);
    const float b5 = fmaf(t, b6, s[5] - b7);
    const float b4 = fmaf(t, b5, s[4] - b6);
    const float b3 = fmaf(t, b4, s[3] - b5);
    const float b2 = fmaf(t, b3, s[2] - b4);
    const float b1 = fmaf(t, b2, s[1] - b3);
    return fmaf(x, b1, s[0] - b2);
}

__device__ __forceinline__ float kan_chain(float x, const float s0[8],
                                           const float s1[8], const float s2[8]) {
    return cheb_eval8(cheb_eval8(cheb_eval8(x, s0), s1), s2);
}

// ---------------------------------------------------------------------------
// Kernel 2: streaming forward pass. float4 vector loads/stores (512B per
// wave32 per access pair), 8 wave32s per 256-thread block.
// ---------------------------------------------------------------------------
__global__ __launch_bounds__(256) void kan_forward(
    const float* __restrict__ x, const float* __restrict__ sums,
    float* __restrict__ out, int n4, int n)
{
    float s0[8], s1[8], s2[8];
    #pragma unroll
    for (int d = 0; d < 8; ++d) {   // 24 floats, L2-resident broadcast
        s0[d] = sums[d];
        s1[d] = sums[8 + d];
        s2[d] = sums[16 + d];
    }

    const int i = blockIdx.x * blockDim.x + threadIdx.x;
    if (i < n4) {
        float4 xv = reinterpret_cast<const float4*>(x)[i];
        float4 yv;
        yv.x = kan_chain(xv.x, s0, s1, s2);
        yv.y = kan_chain(xv.y, s0, s1, s2);
        yv.z = kan_chain(xv.z, s0, s1, s2);
        yv.w = kan_chain(xv.w, s0, s1, s2);
        reinterpret_cast<float4*>(out)[i] = yv;
    }
    // Scalar tail (N is 2^22 so normally empty).
    if (i == 0) {
        for (int j = n4 * 4; j < n; ++j)
            out[j] = kan_chain(x[j], s0, s1, s2);
    }
}

extern "C" void kernel_launch(void* const* d_in, const int* in_sizes, int n_in,
                              void* d_out, int out_size, void* d_ws, size_t ws_size,
                              hipStream_t stream) {
    const float* x  = (const float*)d_in[0];
    const float* c0 = (const float*)d_in[1];
    const float* c1 = (const float*)d_in[2];
    const float* c2 = (const float*)d_in[3];
    float* out  = (float*)d_out;
    float* sums = (float*)d_ws;           // 24 floats of scratch

    const int r0 = in_sizes[1] / 8;       // 128
    const int r1 = in_sizes[2] / 8;       // 128
    const int r2 = in_sizes[3] / 8;       // 1

    kan_colsum_wmma<<<1, 32, 0, stream>>>(c0, c1, c2, r0, r1, r2, sums);

    const int n  = out_size;              // 4194304
    const int n4 = n >> 2;
    const int blocks = (n4 + 255) / 256;  // 4096 blocks
    kan_forward<<<blocks, 256, 0, stream>>>(x, sums, out, n4, n);
}